// PathRefiner_16243566313705
// MI455X (gfx1250) — compile-verified
//
#include <hip/hip_runtime.h>
#include <hip/hip_bf16.h>
#include <math.h>

typedef __attribute__((ext_vector_type(2))) float v2f;
typedef __attribute__((ext_vector_type(8))) float v8f;

#define HF 128
#define WF 128
#define HW (HF * WF)          // 16384
#define CIN 256
#define DD 128
#define TT 16384
#define BB 4
#define MAXD (0.5f / 512.0f)
#define FLD 130               // LDS row stride (pad to avoid bank conflicts)

// ---------------------------------------------------------------------------
// Kernel 1: vm_t[b][p][d] = sum_c fmap[b][c][p] * W_proj[d][c]
// GEMM per batch: M = p (16384), N = d (128), K = c (256), f32 WMMA 16x16x4.
// One wave computes a 16(p) x 128(d) slab: the A-fragment (fmap stream, the
// HBM-bound operand) is loaded once per K-step and reused across 8 WMMAs into
// 8 independent accumulators; B-fragments are immediate-offset loads off one
// base pointer (W_proj is L2/WGP$-resident broadcast data).
// ---------------------------------------------------------------------------
__global__ __launch_bounds__(256) void proj_kernel(
    const float* __restrict__ fmap,   // (B, C, HW)
    const float* __restrict__ Wproj,  // (D, C)
    float* __restrict__ vmt)          // (B, HW, D)
{
    const int tid  = threadIdx.x;
    const int wave = tid >> 5;
    const int lane = tid & 31;

    const int g  = blockIdx.x * 8 + wave;    // 4096 wave-slabs total
    const int b  = g >> 10;                  // 1024 p-tiles per batch
    const int p0 = (g & 1023) << 4;

    const int m  = lane & 15;                // A row (p) == B col (d) lane slot
    const int hi = lane >> 4;
    const int kb = hi << 1;                  // K sub-offset {0,2}

    const float* Ab = fmap + (size_t)b * CIN * HW + p0 + m;   // + c*HW
    const float* Bb = Wproj + (size_t)m * CIN + kb;           // + nt*16*CIN + c

    v8f acc[8] = {};
#pragma unroll 2
    for (int k = 0; k < CIN; k += 4) {
        v2f a;
        a.x = Ab[(size_t)(k + kb) * HW];          // fmap[c=k+kb][p0+m]
        a.y = Ab[(size_t)(k + kb + 1) * HW];
#pragma unroll
        for (int nt = 0; nt < 8; ++nt) {
            v2f bf;
            bf.x = Bb[nt * 16 * CIN + k];         // Wproj[nt*16+m][k+kb]
            bf.y = Bb[nt * 16 * CIN + k + 1];
            acc[nt] = __builtin_amdgcn_wmma_f32_16x16x4_f32(
                false, a, false, bf, (short)0, acc[nt], false, false);
        }
    }

    float* out = vmt + ((size_t)b * HW + p0) * DD + m;        // N = lane&15
#pragma unroll
    for (int r = 0; r < 8; ++r) {
#pragma unroll
        for (int nt = 0; nt < 8; ++nt)
            out[(size_t)(r + 8 * hi) * DD + nt * 16] = acc[nt][r];  // M = r+8*hi
    }
}

// ---------------------------------------------------------------------------
// Kernel 2 (fused): separable 4x4 stencil gather -> feat (LDS) ->
// h = relu(feat @ W1^T + b1) via WMMA -> delta = tanh(h @ W2^T + b2)*MAXD ->
// out = coords + delta.  One block = 16 points of one batch.
// ---------------------------------------------------------------------------
__global__ __launch_bounds__(256) void sample_mlp_kernel(
    const float* __restrict__ vmt,     // (B, HW, D)
    const float* __restrict__ coords,  // (B, T, 2)
    const float* __restrict__ W1,      // (D, D)
    const float* __restrict__ b1,      // (D)
    const float* __restrict__ W2,      // (2, D)
    const float* __restrict__ b2,      // (2)
    float* __restrict__ outp)          // (B, T, 2)
{
    __shared__ float featS[16 * FLD];
    __shared__ float hS[16 * FLD];

    const int tid = threadIdx.x;
    const int b   = blockIdx.x >> 10;          // 1024 tiles per batch
    const int t0  = (blockIdx.x & 1023) << 4;

    // ---- Phase 1: gather. thread = (point tl, channel lane cg), 8 ch each.
    {
        const int tl = tid >> 4;
        const int cg = tid & 15;
        const int tg = t0 + tl;
        const float cx = coords[((size_t)b * TT + tg) * 2 + 0];
        const float cy = coords[((size_t)b * TT + tg) * 2 + 1];
        // integer pixel offsets => shared fractional weights for all 9 samples
        const float ix = cx * (float)(WF - 1);
        const float iy = cy * (float)(HF - 1);
        const float x0f = floorf(ix), y0f = floorf(iy);
        const float fx = ix - x0f, fy = iy - y0f;
        const int x0 = (int)x0f, y0 = (int)y0f;
        const float cw[4] = {1.f - fx, 1.f, 1.f, fx};
        const float rw[4] = {1.f - fy, 1.f, 1.f, fy};

        float acc[8];
#pragma unroll
        for (int c = 0; c < 8; ++c) acc[c] = 0.f;

        const float* vb = vmt + (size_t)b * HW * DD;
#pragma unroll
        for (int j = 0; j < 4; ++j) {
            const int y = y0 - 1 + j;
            if (y < 0 || y >= HF) continue;
#pragma unroll
            for (int i = 0; i < 4; ++i) {
                const int x = x0 - 1 + i;
                if (x < 0 || x >= WF) continue;
                const float w = cw[i] * rw[j];
                const float* pv = vb + ((size_t)y * WF + x) * DD + cg;
#pragma unroll
                for (int c = 0; c < 8; ++c) acc[c] += w * pv[c * 16];
            }
        }
        const float inv9 = 1.0f / 9.0f;
#pragma unroll
        for (int c = 0; c < 8; ++c)
            featS[tl * FLD + cg + c * 16] = acc[c] * inv9;
    }
    __syncthreads();

    // ---- Phase 2: h = relu(feat(16xD) @ W1^T + b1), wave w -> N-tile w*16.
    const int wave = tid >> 5;
    const int lane = tid & 31;
    const int m  = lane & 15;       // A row (point) == B col (out channel)
    const int hi = lane >> 4;
    const int kb = hi << 1;
    const int n0 = wave << 4;

    v8f c8 = {};
    const float* w1p = W1 + (size_t)(n0 + m) * DD;   // B[k][n] = W1[n][k]
#pragma unroll 4
    for (int k = 0; k < DD; k += 4) {
        v2f a, bf;
        a.x  = featS[m * FLD + k + kb];
        a.y  = featS[m * FLD + k + kb + 1];
        bf.x = w1p[k + kb];
        bf.y = w1p[k + kb + 1];
        c8 = __builtin_amdgcn_wmma_f32_16x16x4_f32(
            false, a, false, bf, (short)0, c8, false, false);
    }
    const float bias = b1[n0 + m];
#pragma unroll
    for (int r = 0; r < 8; ++r) {
        const float v = c8[r] + bias;                // element (M=r+8*hi, N=m)
        hS[(r + 8 * hi) * FLD + n0 + m] = v > 0.f ? v : 0.f;
    }
    __syncthreads();

    // ---- Phase 3: tiny 16x2 head + tanh + output (wave 0).
    if (wave == 0) {
        const int t = lane & 15;
        const int j = lane >> 4;                     // output component 0/1
        float dot = b2[j];
        const float* w2p = W2 + (size_t)j * DD;
#pragma unroll 8
        for (int k = 0; k < DD; ++k) dot += hS[t * FLD + k] * w2p[k];
        const float delta = tanhf(dot) * MAXD;
        const size_t oi = ((size_t)b * TT + (t0 + t)) * 2 + j;
        outp[oi] = coords[oi] + delta;
    }
}

extern "C" void kernel_launch(void* const* d_in, const int* in_sizes, int n_in,
                              void* d_out, int out_size, void* d_ws, size_t ws_size,
                              hipStream_t stream) {
    const float* fmap   = (const float*)d_in[0];  // (4,256,128,128)
    const float* coords = (const float*)d_in[1];  // (4,16384,2)
    const float* Wproj  = (const float*)d_in[2];  // (128,256)
    const float* W1     = (const float*)d_in[3];  // (128,128)
    const float* b1     = (const float*)d_in[4];  // (128)
    const float* W2     = (const float*)d_in[5];  // (2,128)
    const float* b2     = (const float*)d_in[6];  // (2)
    float* out = (float*)d_out;

    float* vmt = (float*)d_ws;                    // 4*16384*128*4 = 32 MB

    proj_kernel<<<512, 256, 0, stream>>>(fmap, Wproj, vmt);
    sample_mlp_kernel<<<4096, 256, 0, stream>>>(vmt, coords, W1, b1, W2, b2, out);
}